// CRF_1254130451065
// MI455X (gfx1250) — compile-verified
//
#include <hip/hip_runtime.h>

typedef __attribute__((ext_vector_type(2))) float v2f;
typedef __attribute__((ext_vector_type(4))) float v4f;
typedef __attribute__((ext_vector_type(8))) float v8f;

#define B_ 64
#define T_ 512
#define D_ 1024
#define U_ 128
#define KCHUNK 64
#define NPAIRS (KCHUNK / 2)   // 32 K-pairs per chunk
#define SROW 144              // LDS pair-row stride in v2f units (bank-stagger: 2*144 mod 64 == 32)

// energy[b,t,u] = sum_d x[b,t,d]*kernel[d,u] + bias[u]
//              + start_mask(b,t)*left[u] + end_mask(b,t)*right[u]
__global__ __launch_bounds__(256) void crf_energy_wmma(
    const float* __restrict__ x,      // [B*T, D]
    const int*   __restrict__ mask,   // [B*T]
    const float* __restrict__ kern,   // [D, U]
    const float* __restrict__ bias,   // [U]
    const float* __restrict__ lbnd,   // [U]
    const float* __restrict__ rbnd,   // [U]
    float*       __restrict__ out)    // [B*T, U]
{
    __shared__ v2f ldsB[NPAIRS * SROW];  // [pair p][n] -> (kern[2p][n], kern[2p+1][n])

    const int tid  = threadIdx.x;
    const int lane = tid & 31;
    const int wv   = tid >> 5;
    const int half = lane >> 4;      // 0 = lanes 0-15, 1 = lanes 16-31
    const int l16  = lane & 15;

    const int mbase = (blockIdx.x * 8 + wv) * 16;        // this wave's 16-row tile
    const float* xrow = x + (long)(mbase + l16) * D_;    // A: M = lane%16

    const v8f vzero = {0.f, 0.f, 0.f, 0.f, 0.f, 0.f, 0.f, 0.f};
    v8f acc[8];
#pragma unroll
    for (int i = 0; i < 8; ++i) acc[i] = vzero;

    for (int kc = 0; kc < D_; kc += KCHUNK) {
        __syncthreads();
        // Cooperatively stage kernel chunk [kc, kc+KCHUNK) x U into LDS as K-pairs.
        // Each thread stages 4 quad-groups: pair p, columns n0..n0+3.
        //   2 x global_load_b128 (coalesced: a wave covers one full 512B kernel row)
        //   2 x ds_store_b128    (32B contiguous, 16B aligned: SROW*8 = 1152 = 36*32)
#pragma unroll
        for (int i = 0; i < (NPAIRS * (U_ / 4)) / 256; ++i) {  // 4 iterations
            int Q  = tid + i * 256;
            int p  = Q >> 5;            // pair index 0..31
            int n0 = (Q & 31) << 2;     // column 0,4,...,124
            int kg = kc + 2 * p;
            v4f g0 = *(const v4f*)(kern + (long)kg * U_ + n0);
            v4f g1 = *(const v4f*)(kern + (long)(kg + 1) * U_ + n0);
            v4f lo = {g0.x, g1.x, g0.y, g1.y};   // pairs for n0, n0+1
            v4f hi = {g0.z, g1.z, g0.w, g1.w};   // pairs for n0+2, n0+3
            v2f* dst = &ldsB[p * SROW + n0];
            *(v4f*)(dst)     = lo;
            *(v4f*)(dst + 2) = hi;
        }
        __syncthreads();

        if (kc + KCHUNK < D_) {
            __builtin_prefetch(xrow + kc + KCHUNK, 0, 1);  // next A chunk
        }

#pragma unroll
        for (int ks = 0; ks < KCHUNK; ks += 4) {
            // A fragment (16x4 f32): lanes 0-15 -> K=k0,k0+1 ; lanes 16-31 -> K=k0+2,k0+3
            int ka = kc + ks + half * 2;
            v2f a = *(const v2f*)(xrow + ka);
            int p = (ks >> 1) + half;   // B pair row for this half
#pragma unroll
            for (int nt = 0; nt < 8; ++nt) {
                // B fragment (4x16 f32): lane holds (K[k0+2h][n], K[k0+2h+1][n])
                v2f b = ldsB[p * SROW + nt * 16 + l16];
                acc[nt] = __builtin_amdgcn_wmma_f32_16x16x4_f32(
                    false, a, false, b, (short)0, acc[nt], false, false);
            }
        }
    }

    // Per-row boundary flags (rows this lane owns: mbase + r + 8*half, r=0..7).
    float sf[8], ef[8];
#pragma unroll
    for (int r = 0; r < 8; ++r) {
        int row = mbase + r + half * 8;
        int t   = row & (T_ - 1);
        int mc  = mask[row];
        int mp  = (t > 0)      ? mask[row - 1] : 0;  // shift_right pads 0 at front
        int mn  = (t < T_ - 1) ? mask[row + 1] : 0;  // shift_left pads 0 at end
        sf[r] = (mc > mp) ? 1.0f : 0.0f;
        ef[r] = (mn > mc) ? 1.0f : 0.0f;
    }

    // Epilogue: bias + boundary energies, then store.
#pragma unroll
    for (int nt = 0; nt < 8; ++nt) {
        int c = nt * 16 + l16;
        float bv = bias[c];
        float lv = lbnd[c];
        float rv = rbnd[c];
#pragma unroll
        for (int r = 0; r < 8; ++r) {
            int row = mbase + r + half * 8;
            float v = acc[nt][r] + bv + sf[r] * lv + ef[r] * rv;
            out[(long)row * U_ + c] = v;
        }
    }
}

extern "C" void kernel_launch(void* const* d_in, const int* in_sizes, int n_in,
                              void* d_out, int out_size, void* d_ws, size_t ws_size,
                              hipStream_t stream) {
    const float* x    = (const float*)d_in[0];
    const int*   mask = (const int*)d_in[1];
    const float* kern = (const float*)d_in[2];
    const float* bias = (const float*)d_in[3];
    const float* lbnd = (const float*)d_in[4];
    const float* rbnd = (const float*)d_in[5];
    float* out = (float*)d_out;

    // M = B*T = 32768 rows; 8 waves * 16 rows = 128 rows per block -> 256 blocks.
    dim3 grid((B_ * T_) / 128);
    dim3 block(256);
    hipLaunchKernelGGL(crf_energy_wmma, grid, block, 0, stream,
                       x, mask, kern, bias, lbnd, rbnd, out);
}